// CamEmbedding_14937896255960
// MI455X (gfx1250) — compile-verified
//
#include <hip/hip_runtime.h>
#include <hip/hip_bf16.h>

#define NUM_EXPERTS 8
#define DIM 2048
#define BATCH 8192

typedef __attribute__((ext_vector_type(2))) float v2f;
typedef __attribute__((ext_vector_type(4))) float v4f;
typedef __attribute__((ext_vector_type(8))) float v8f;

// -------- workspace layout (ints) --------
// [0..7]    counts per expert
// [8..16]   exclusive-scan offsets (9 entries)
// [20..27]  atomic cursors
// [32..32+BATCH-1] permutation (token ids grouped by expert)
#define WS_COUNTS  0
#define WS_OFFSETS 8
#define WS_CURSORS 20
#define WS_PERM    32

__global__ void moe_init(int* ws) {
    int t = threadIdx.x;
    if (t < NUM_EXPERTS) {
        ws[WS_COUNTS + t]  = 0;
        ws[WS_CURSORS + t] = 0;
    }
}

__global__ void moe_hist(const long long* __restrict__ ids, int* ws) {
    int i = blockIdx.x * blockDim.x + threadIdx.x;
    if (i < BATCH) {
        int e = (int)ids[i];
        atomicAdd(&ws[WS_COUNTS + e], 1);
    }
}

__global__ void moe_scan(int* ws) {
    if (threadIdx.x == 0) {
        int acc = 0;
        for (int e = 0; e < NUM_EXPERTS; ++e) {
            ws[WS_OFFSETS + e] = acc;
            acc += ws[WS_COUNTS + e];
        }
        ws[WS_OFFSETS + NUM_EXPERTS] = acc;
    }
}

__global__ void moe_scatter(const long long* __restrict__ ids, int* ws) {
    int i = blockIdx.x * blockDim.x + threadIdx.x;
    if (i < BATCH) {
        int e = (int)ids[i];
        int pos = atomicAdd(&ws[WS_CURSORS + e], 1);
        ws[WS_PERM + ws[WS_OFFSETS + e] + pos] = i;
    }
}

#define WMMA_F32(A, B, C) \
    __builtin_amdgcn_wmma_f32_16x16x4_f32(false, (A), false, (B), (short)0, (C), false, false)

// Grouped GEMM: out[row] = x[row] + relu(x[row] @ W[e]^T + b[e])
// Block tile: 128 rows x 128 cols. 8 waves as 4(row) x 2(col);
// each wave: 32 rows (2 M-subtiles) x 64 cols (4 N-subtiles)
// -> 8 accumulators of v8f, V_WMMA_F32_16X16X4_F32 over K=2048.
//
// K mapping trick: lane group g (= lane>>4) loads 4 contiguous floats at
// k + 4g with one b128 load. WMMA "lo" uses .xy (K = k+4g, k+4g+1), WMMA
// "hi" uses .zw (K = k+4g+2, k+4g+3). A and B use the same mapping, and
// across both lane groups the two WMMAs cover K = k..k+7 exactly once,
// so accumulation is identical to the canonical layout — but one 128-bit
// load now feeds two WMMAs (6 loads per 16 WMMAs).
__global__ __launch_bounds__(256, 2)
void moe_gemm(const float* __restrict__ x,
              const float* __restrict__ W,
              const float* __restrict__ bias,
              const int* __restrict__ ws,
              float* __restrict__ out) {
    __shared__ int rowIdx[128];

    // Map blockIdx.x -> (expert, rowTile) by scanning per-expert tile counts.
    const int bid = blockIdx.x;
    int expert = -1, rowTile = 0, cnt = 0, off = 0, acc = 0;
    for (int e = 0; e < NUM_EXPERTS; ++e) {
        int c  = ws[WS_COUNTS + e];
        int nt = (c + 127) >> 7;
        if (expert < 0 && bid < acc + nt) {
            expert  = e;
            rowTile = bid - acc;
            cnt     = c;
            off     = ws[WS_OFFSETS + e];
        }
        acc += nt;
    }
    if (expert < 0) return;   // uniform per block: all threads exit together

    const int tid = threadIdx.x;
    if (tid < 128) {
        int r = rowTile * 128 + tid;
        rowIdx[tid] = (r < cnt) ? ws[WS_PERM + off + r] : -1;
    }
    __syncthreads();

    const int lane = tid & 31;
    const int wave = tid >> 5;
    const int wr = wave >> 1;       // 0..3  row wave
    const int wc = wave & 1;        // 0..1  col wave
    const int ln = lane & 15;       // N (B/C/D) or M (A) within subtile
    const int kh = lane >> 4;       // lane group g

    const int colBase = blockIdx.y * 128 + wc * 64;

    // A fragment pointers: lane holds row (M = ln), b128 at k + 4*kh
    const float* aptr[2];
    #pragma unroll
    for (int s = 0; s < 2; ++s) {
        int rl = wr * 32 + s * 16 + ln;
        int ri = rowIdx[rl];
        aptr[s] = x + (size_t)(ri < 0 ? 0 : ri) * DIM + 4 * kh;
    }
    // B fragment pointers: B[k][n] = W[e][n][k] (row-major W == col-major B)
    const float* bptr[4];
    #pragma unroll
    for (int t = 0; t < 4; ++t) {
        int col = colBase + t * 16 + ln;
        bptr[t] = W + ((size_t)expert * DIM + col) * DIM + 4 * kh;
    }

    v8f c[2][4] = {};

    for (int k = 0; k < DIM; k += 8) {
        v4f a0 = *(const v4f*)(aptr[0] + k);
        v4f a1 = *(const v4f*)(aptr[1] + k);
        v4f b0 = *(const v4f*)(bptr[0] + k);
        v4f b1 = *(const v4f*)(bptr[1] + k);
        v4f b2 = *(const v4f*)(bptr[2] + k);
        v4f b3 = *(const v4f*)(bptr[3] + k);

        v2f a0lo = {a0.x, a0.y}, a0hi = {a0.z, a0.w};
        v2f a1lo = {a1.x, a1.y}, a1hi = {a1.z, a1.w};
        v2f b0lo = {b0.x, b0.y}, b0hi = {b0.z, b0.w};
        v2f b1lo = {b1.x, b1.y}, b1hi = {b1.z, b1.w};
        v2f b2lo = {b2.x, b2.y}, b2hi = {b2.z, b2.w};
        v2f b3lo = {b3.x, b3.y}, b3hi = {b3.z, b3.w};

        // lo sweep (K = k+4g .. k+4g+1)
        c[0][0] = WMMA_F32(a0lo, b0lo, c[0][0]);
        c[0][1] = WMMA_F32(a0lo, b1lo, c[0][1]);
        c[0][2] = WMMA_F32(a0lo, b2lo, c[0][2]);
        c[0][3] = WMMA_F32(a0lo, b3lo, c[0][3]);
        c[1][0] = WMMA_F32(a1lo, b0lo, c[1][0]);
        c[1][1] = WMMA_F32(a1lo, b1lo, c[1][1]);
        c[1][2] = WMMA_F32(a1lo, b2lo, c[1][2]);
        c[1][3] = WMMA_F32(a1lo, b3lo, c[1][3]);
        // hi sweep (K = k+4g+2 .. k+4g+3)
        c[0][0] = WMMA_F32(a0hi, b0hi, c[0][0]);
        c[0][1] = WMMA_F32(a0hi, b1hi, c[0][1]);
        c[0][2] = WMMA_F32(a0hi, b2hi, c[0][2]);
        c[0][3] = WMMA_F32(a0hi, b3hi, c[0][3]);
        c[1][0] = WMMA_F32(a1hi, b0hi, c[1][0]);
        c[1][1] = WMMA_F32(a1hi, b1hi, c[1][1]);
        c[1][2] = WMMA_F32(a1hi, b2hi, c[1][2]);
        c[1][3] = WMMA_F32(a1hi, b3hi, c[1][3]);
    }

    // Bias per lane (depends only on column)
    float bv[4];
    #pragma unroll
    for (int t = 0; t < 4; ++t)
        bv[t] = bias[expert * DIM + colBase + t * 16 + ln];

    // C/D layout: VGPR v -> M = kh*8 + v (lanes 0-15: rows 0-7; 16-31: rows 8-15), N = ln
    #pragma unroll
    for (int s = 0; s < 2; ++s) {
        #pragma unroll
        for (int v = 0; v < 8; ++v) {
            int rl = wr * 32 + s * 16 + kh * 8 + v;
            int ri = rowIdx[rl];
            if (ri < 0) continue;
            #pragma unroll
            for (int t = 0; t < 4; ++t) {
                int col = colBase + t * 16 + ln;
                float val = c[s][t][v] + bv[t];
                val = val > 0.0f ? val : 0.0f;
                size_t idx = (size_t)ri * DIM + col;
                out[idx] = x[idx] + val;
            }
        }
    }
}

extern "C" void kernel_launch(void* const* d_in, const int* in_sizes, int n_in,
                              void* d_out, int out_size, void* d_ws, size_t ws_size,
                              hipStream_t stream) {
    const float*     x    = (const float*)d_in[0];
    const long long* ids  = (const long long*)d_in[1];
    const float*     W    = (const float*)d_in[2];
    const float*     bias = (const float*)d_in[3];
    float*           out  = (float*)d_out;
    int*             ws   = (int*)d_ws;

    moe_init<<<1, 32, 0, stream>>>(ws);
    moe_hist<<<BATCH / 256, 256, 0, stream>>>(ids, ws);
    moe_scan<<<1, 32, 0, stream>>>(ws);
    moe_scatter<<<BATCH / 256, 256, 0, stream>>>(ids, ws);

    // Max (expert,rowTile) pairs: BATCH/128 full tiles + up to 8 partial tails.
    const int maxRowTiles = BATCH / 128 + NUM_EXPERTS;   // 72
    dim3 grid(maxRowTiles, DIM / 128);                   // 72 x 16
    moe_gemm<<<grid, 256, 0, stream>>>(x, W, bias, ws, out);
}